// SpatialTransformer_37477884625242
// MI455X (gfx1250) — compile-verified
//
#include <hip/hip_runtime.h>
#include <cstdint>

// Problem geometry (fixed by the reference)
#define NB 2
#define SX 160
#define SY 192
#define SZ 160
#define NVOX (NB * SX * SY * SZ)     // 9,830,400
#define VOXB (SX * SY * SZ)          // 4,915,200 per batch
#define TPB 256
#define VPT 4                         // voxels per thread (share one (b,x,y) row)
#define TILE (TPB * VPT)              // 1024 voxels / tile
#define NTILES (NVOX / TILE)          // 9600 exactly
#define TRF_TILE_FLOATS (TILE * 3)    // 3072 floats = 12 KB per buffer
#define GRID_BLOCKS 1536              // ~6.25 tiles per block -> pipeline overlap

typedef unsigned int uint32;
typedef uint32 v4u __attribute__((ext_vector_type(4)));
typedef int    v4i __attribute__((ext_vector_type(4)));
typedef int    v8i __attribute__((ext_vector_type(8)));

// Issue one TDM DMA: copy `ndwords` dwords from global `gsrc` into LDS at byte
// offset `ldsByteOff`. 1-D tensor descriptor, data_size = 4B, type = 2 (image).
// Tracked by TENSORcnt; completion in issue order within the wave.
// clang-23 / therock-10.0 6-arg builtin form:
//   (uint32x4 g0, int32x8 g1, int32x4 g2, int32x4 g3, int32x8 g4, i32 cpol)
__device__ __forceinline__ void tdm_load_tile(const float* gsrc, uint32 ldsByteOff,
                                              int ndwords) {
  uint64_t ga = (uint64_t)(uintptr_t)gsrc;
  v4u g0;
  g0.x = 1u;                                   // count=1, user descriptor, no gather
  g0.y = ldsByteOff;                           // lds_addr (bytes)
  g0.z = (uint32)(ga & 0xFFFFFFFFu);           // global_addr[31:0]
  g0.w = (uint32)((ga >> 32) & 0x01FFFFFFu)    // global_addr[56:32]
       | (2u << 30);                           // type = 2
  v8i g1;
  g1[0] = (int)(2u << 16);                     // workgroup_mask=0, data_size=2 (4B)
  g1[1] = (int)(((uint32)ndwords & 0xFFFFu) << 16);  // tensor_dim0[15:0] @ bits 63:48
  g1[2] = (int)((uint32)ndwords >> 16);              // tensor_dim0[31:16]; tensor_dim1=0
  g1[3] = (int)(((uint32)ndwords & 0xFFFFu) << 16);  // tile_dim0 @ bits 127:112
  g1[4] = 0;                                   // tile_dim1=0, tile_dim2=0
  g1[5] = 0; g1[6] = 0; g1[7] = 0;             // strides unused for 1-D tile
  v4i z4 = {0, 0, 0, 0};                       // groups 2/3 disabled (<=2D tensor)
  v8i z8 = {0, 0, 0, 0, 0, 0, 0, 0};           // extra group (unused)
  __builtin_amdgcn_tensor_load_to_lds(g0, g1, z4, z4, z8, 0);
}

__global__ __launch_bounds__(TPB)
void spatial_transformer_warp(const float* __restrict__ vol,
                              const float* __restrict__ trf,
                              float* __restrict__ out) {
  __shared__ __align__(16) float smTrf[2][TRF_TILE_FLOATS];

  const int tid    = threadIdx.x;
  const int stride = (int)gridDim.x;
  const int t0     = (int)blockIdx.x;

  const uint32 lds0 = (uint32)(uintptr_t)(&smTrf[0][0]);
  const uint32 lds1 = (uint32)(uintptr_t)(&smTrf[1][0]);

  // Prologue: prime the double buffer with up to two tiles.
  if (tid == 0) {
    if (t0 < NTILES)
      tdm_load_tile(trf + (size_t)t0 * TILE * 3, lds0, TRF_TILE_FLOATS);
    if (t0 + stride < NTILES)
      tdm_load_tile(trf + (size_t)(t0 + stride) * TILE * 3, lds1, TRF_TILE_FLOATS);
  }

  int parity = 0;
  for (int t = t0; t < NTILES; t += stride, parity ^= 1) {
    // Wait for this tile's DMA. If a prefetch for t+stride is also in flight,
    // waiting for TENSORcnt<=1 releases us as soon as the older op lands.
    if (tid == 0) {
      if (t + stride < NTILES) __builtin_amdgcn_s_wait_tensorcnt(1);
      else                     __builtin_amdgcn_s_wait_tensorcnt(0);
    }
    __syncthreads();  // publish LDS tile to all 8 waves

    const float* sbuf = &smTrf[parity][0];
    const int base = t * TILE + tid * VPT;   // flat voxel base for this thread

    // 12 staged floats per thread: 3x ds_load_b128, 48B stride (bank-conflict-free)
    const float4* s4 = (const float4*)(sbuf + tid * 12);
    float4 da = s4[0], db = s4[1], dc = s4[2];
    float d[12] = {da.x, da.y, da.z, da.w,
                   db.x, db.y, db.z, db.w,
                   dc.x, dc.y, dc.z, dc.w};

    // base is a multiple of 4 and SZ % 4 == 0, so the thread's 4 voxels share
    // (b, x, y) and have z = zq + j (no row wrap). Decode once.
    const int zq = base % SZ;  const int r1 = base / SZ;
    const int yq = r1 % SY;    const int r2 = r1 / SY;
    const int xq = r2 % SX;    const int bq = r2 / SX;
    const float fxq = (float)xq;
    const float fyq = (float)yq;
    const float fzq = (float)zq;
    const int   ibb = bq * VOXB;

    float4 res;
    float* rp = &res.x;
    #pragma unroll
    for (int j = 0; j < VPT; ++j) {
      // Sample location, clamped to the volume border (voxelmorph semantics)
      const float lx = __builtin_amdgcn_fmed3f(fxq + d[3 * j + 0], 0.0f, (float)(SX - 1));
      const float ly = __builtin_amdgcn_fmed3f(fyq + d[3 * j + 1], 0.0f, (float)(SY - 1));
      const float lz = __builtin_amdgcn_fmed3f((fzq + (float)j) + d[3 * j + 2],
                                               0.0f, (float)(SZ - 1));

      // lx,ly,lz >= 0 so trunc == floor; fract gives the lerp weight directly
      const int x0 = (int)lx, y0 = (int)ly, z0 = (int)lz;
      const float fx = __builtin_amdgcn_fractf(lx);
      const float fy = __builtin_amdgcn_fractf(ly);
      const float fz = __builtin_amdgcn_fractf(lz);

      // Border-clamped corner strides (upper corner collapses onto lower at max)
      const int dx  = (x0 < SX - 1) ? (SY * SZ) : 0;
      const int dy  = (y0 < SY - 1) ? SZ : 0;
      const int dzi = (z0 < SZ - 1) ? 1 : 0;
      const int i000 = ibb + x0 * (SY * SZ) + y0 * SZ + z0;   // 32-bit indices

      const float v000 = vol[i000];
      const float v001 = vol[i000 + dzi];
      const float v010 = vol[i000 + dy];
      const float v011 = vol[i000 + dy + dzi];
      const float v100 = vol[i000 + dx];
      const float v101 = vol[i000 + dx + dzi];
      const float v110 = vol[i000 + dx + dy];
      const float v111 = vol[i000 + dx + dy + dzi];

      const float c00 = v000 + fz * (v001 - v000);
      const float c01 = v010 + fz * (v011 - v010);
      const float c10 = v100 + fz * (v101 - v100);
      const float c11 = v110 + fz * (v111 - v110);
      const float c0  = c00 + fy * (c01 - c00);
      const float c1  = c10 + fy * (c11 - c10);
      rp[j] = c0 + fx * (c1 - c0);
    }

    *(float4*)(out + base) = res;   // global_store_b128, fully coalesced

    __syncthreads();  // all waves done reading smTrf[parity]
    // Prefetch tile t+2*stride into the buffer we just drained.
    if (tid == 0 && (t + 2 * stride) < NTILES) {
      tdm_load_tile(trf + (size_t)(t + 2 * stride) * TILE * 3,
                    parity ? lds1 : lds0, TRF_TILE_FLOATS);
    }
  }
}

extern "C" void kernel_launch(void* const* d_in, const int* in_sizes, int n_in,
                              void* d_out, int out_size, void* d_ws, size_t ws_size,
                              hipStream_t stream) {
  const float* vol = (const float*)d_in[0];   // [2,160,192,160,1] f32
  const float* trf = (const float*)d_in[1];   // [2,160,192,160,3] f32
  float* out = (float*)d_out;                 // [2,160,192,160,1] f32
  (void)in_sizes; (void)n_in; (void)out_size; (void)d_ws; (void)ws_size;

  spatial_transformer_warp<<<GRID_BLOCKS, TPB, 0, stream>>>(vol, trf, out);
}